// particle_flow_5987184410827
// MI455X (gfx1250) — compile-verified
//
#include <hip/hip_runtime.h>
#include <cmath>
#include <cstdint>

#define DX 784
#define DZ 128
#define DH 8192
#define KB 64            // K-block staged into LDS per TDM transfer

typedef __attribute__((ext_vector_type(2))) float v2f;
typedef __attribute__((ext_vector_type(8))) float v8f;

#ifndef __has_builtin
#define __has_builtin(x) 0
#endif
#if __has_builtin(__builtin_amdgcn_tensor_load_to_lds) && \
    __has_builtin(__builtin_amdgcn_s_wait_tensorcnt)
#define HAVE_TDM 1
#else
#define HAVE_TDM 0
#endif

// -------- fp32 WMMA wrapper: D(16x16) += A(16x4) * B(4x16) ----------------
__device__ __forceinline__ v8f wmma_f32_k4(v2f a, v2f b, v8f c) {
  // 8 args: (neg_a, A, neg_b, B, c_mod, C, reuse_a, reuse_b)
  return __builtin_amdgcn_wmma_f32_16x16x4_f32(false, a, false, b, (short)0, c,
                                               false, false);
}

#if HAVE_TDM
typedef __attribute__((ext_vector_type(4))) unsigned tdm_u4;
typedef __attribute__((ext_vector_type(4))) int tdm_i4;
typedef __attribute__((ext_vector_type(8))) int tdm_i8;

// Issue a TDM 2D-tile load: Wd2 rows [k0, k0+KB) x cols [m0, m0+16) -> LDS.
// D# layout per CDNA5 ISA ch.8 (group0: count/lds/global/type, group1: dims).
__device__ __forceinline__ void tdm_issue_a_panel(const float* gptr,
                                                  unsigned lds_off) {
  unsigned long long ga = (unsigned long long)(uintptr_t)gptr;
  tdm_u4 g0;
  g0.x = 1u;                                           // count=1 (valid user D#)
  g0.y = lds_off;                                      // lds_addr (bytes)
  g0.z = (unsigned)(ga & 0xFFFFFFFFull);               // global_addr[31:0]
  g0.w = (unsigned)((ga >> 32) & 0x01FFFFFFull)        // global_addr[56:32]
         | (2u << 30);                                 // type = 2 ("image")
  tdm_i8 g1;
  g1[0] = (int)(2u << 16);                             // wg_mask=0, data_size=4B
  g1[1] = (int)(((unsigned)DX & 0xFFFFu) << 16);       // tensor_dim0.lo16 = 784
  g1[2] = (int)(((unsigned)DH & 0xFFFFu) << 16);       // dim0.hi16=0 | dim1.lo16
  g1[3] = (int)((16u << 16) | ((unsigned)DH >> 16));   // dim1.hi16 | tile_dim0=16
  g1[4] = (int)(KB & 0xFFFF);                          // tile_dim1=KB, tile_dim2=0
  g1[5] = (int)DX;                                     // tensor_dim0_stride = 784
  g1[6] = 0;                                           // stride0.hi | stride1.lo
  g1[7] = 0;
  tdm_i4 gz = {0, 0, 0, 0};                            // 2D tensor: groups 2/3 = 0
#if __clang_major__ >= 23
  tdm_i8 gz8 = {0, 0, 0, 0, 0, 0, 0, 0};
  __builtin_amdgcn_tensor_load_to_lds(g0, g1, gz, gz, gz8, 0);
#else
  __builtin_amdgcn_tensor_load_to_lds(g0, g1, gz, gz, 0);
#endif
}
#endif

// Cooperative fallback staging (also used by the host-compile pass).
__device__ __forceinline__ void stage_a_panel(float* dst,
                                              const float* __restrict__ Wd2,
                                              int m0, int k0, int tid) {
  for (int e = tid; e < KB * 16; e += 256) {
    int kk = e >> 4, m = e & 15;
    dst[e] = Wd2[(size_t)(k0 + kk) * DX + m0 + m];
  }
}

// ---------------- encoder: h = tanh(x @ We1 + be1) -------------------------
__global__ void pf_encode_h(const float* __restrict__ x,
                            const float* __restrict__ We1,
                            const float* __restrict__ be1,
                            float* __restrict__ h) {
  int j = blockIdx.x * blockDim.x + threadIdx.x;
  if (j >= DH) return;
  float s = be1[j];
  for (int i = 0; i < DX; ++i) s = fmaf(x[i], We1[(size_t)i * DH + j], s);
  h[j] = tanhf(s);
}

// ------- mu / sigma / z / carry init (1 block, 128 threads) ---------------
__global__ void pf_encode_muz(const float* __restrict__ h,
                              const float* __restrict__ Wmu,
                              const float* __restrict__ bmu,
                              const float* __restrict__ Wsig,
                              const float* __restrict__ bsig,
                              const float* __restrict__ eps_std,
                              float* mu, float* sigma, float* sigma2, float* z,
                              float* bar, float* eta1, float* log_alpha) {
  int j = threadIdx.x;
  float sm = bmu[j], ss = bsig[j];
  for (int k = 0; k < DH; ++k) {
    float hk = h[k];
    sm = fmaf(hk, Wmu[(size_t)k * DZ + j], sm);
    ss = fmaf(hk, Wsig[(size_t)k * DZ + j], ss);
  }
  float sp = (ss > 20.f) ? ss : log1pf(expf(ss));   // softplus
  float sg = sp + 1e-3f;
  float s2 = sg * sg;
  float eps = sm + s2 * eps_std[j];
  float zz = sg * eps + sm;
  mu[j] = sm; sigma[j] = sg; sigma2[j] = s2; z[j] = zz;
  bar[j] = sm;          // bar_eta <- mu
  eta1[j] = zz;         // eta_1  <- z
  if (j == 0) *log_alpha = 0.f;
}

// -------- decoder hidden: v = zin @ Wd1 + bd1; t = tanh(v); d = 1 - t^2 ----
__global__ void pf_dec_hidden(const float* __restrict__ zin,
                              const float* __restrict__ Wd1,
                              const float* __restrict__ bd1,
                              float* __restrict__ t, float* __restrict__ d) {
  int j = blockIdx.x * blockDim.x + threadIdx.x;
  if (j >= DH) return;
  float s = bd1[j];
  for (int k = 0; k < DZ; ++k) s = fmaf(zin[k], Wd1[(size_t)k * DH + j], s);
  float tt = tanhf(s);
  t[j] = tt;
  d[j] = 1.f - tt * tt;
}

// -------- decoder output, three modes --------------------------------------
__global__ void pf_dec_out(const float* __restrict__ t,
                           const float* __restrict__ Wd2,
                           const float* __restrict__ bd2,
                           int mode, const float* __restrict__ x,
                           float* sp, float* rinv, float* rinvx, float* xrec) {
  int i = blockIdx.x * blockDim.x + threadIdx.x;
  if (i >= DX) return;
  float s = bd2[i];
  for (int k = 0; k < DH; ++k) s = fmaf(t[k], Wd2[(size_t)k * DX + i], s);
  float xt = 1.f / (1.f + expf(-s));
  if (mode == 0) {
    float r = xt * (1.f - xt);
    rinv[i] = 1.f / r;
    rinvx[i] = x[i] / r;
  } else if (mode == 1) {
    sp[i] = xt * (1.f - xt);
  } else {
    xrec[i] = 1.f / (1.f + expf(-xt));      // reference double-sigmoids
  }
}

// -------- one-time transpose: Wd1T[k*128+j] = Wd1[j*8192+k] ----------------
__global__ void pf_transpose(const float* __restrict__ Wd1,
                             float* __restrict__ Wd1T) {
  int o = blockIdx.x * blockDim.x + threadIdx.x;
  if (o >= DH * DZ) return;
  int k = o / DZ, j = o - k * DZ;
  Wd1T[o] = Wd1[(size_t)j * DH + k];
}

// -------- WMMA GEMM 1: H[i][j] = sp[i] * sum_k Wd2[k][i]*d[k]*Wd1T[k][j] ---
// Block = one 16-row M tile; 8 waves = 8 N tiles.  A panel (shared by all 8
// waves) is DMA'd into LDS by the Tensor Data Mover, double buffered.
__global__ void pf_jacobian(const float* __restrict__ Wd2,
                            const float* __restrict__ Wd1T,
                            const float* __restrict__ dgate,
                            const float* __restrict__ sp,
                            float* __restrict__ H) {
  __shared__ float As[2][KB * 16];
  int tid = threadIdx.x;
  int wv = tid >> 5;                 // wave in block = N tile
  int lane = tid & 31;
  int m0 = blockIdx.x * 16;
  int n0 = wv * 16;
  int half = lane >> 4, l = lane & 15;
  v8f acc = {};

#if HAVE_TDM
  if (wv == 0)
    tdm_issue_a_panel(Wd2 + (size_t)0 * DX + m0,
                      (unsigned)(uintptr_t)(&As[0][0]));
#else
  stage_a_panel(&As[0][0], Wd2, m0, 0, tid);
#endif

  int buf = 0;
  for (int k0 = 0; k0 < DH; k0 += KB) {
#if HAVE_TDM
    if (wv == 0) __builtin_amdgcn_s_wait_tensorcnt(0);  // buf ready (issuer)
    __syncthreads();                                    // publish to block
    if (wv == 0 && k0 + KB < DH)
      tdm_issue_a_panel(Wd2 + (size_t)(k0 + KB) * DX + m0,
                        (unsigned)(uintptr_t)(&As[buf ^ 1][0]));
#else
    __syncthreads();
    if (k0 + KB < DH)
      stage_a_panel(&As[buf ^ 1][0], Wd2, m0, k0 + KB, tid);
#endif
    if (k0 + KB < DH)
      __builtin_prefetch(Wd1T + (size_t)(k0 + KB) * DZ + n0 + l, 0, 0);

    const float* ap = &As[buf][0];
#pragma unroll 4
    for (int kq = 0; kq < KB; kq += 4) {
      int kl = kq + 2 * half;        // local K pair for this half-wave
      int ka = k0 + kl;
      v2f a, b;
      a.x = ap[kl * 16 + l] * dgate[ka];
      a.y = ap[(kl + 1) * 16 + l] * dgate[ka + 1];
      b.x = Wd1T[(size_t)ka * DZ + n0 + l];
      b.y = Wd1T[(size_t)(ka + 1) * DZ + n0 + l];
      acc = wmma_f32_k4(a, b, acc);
    }
    __syncthreads();                 // done reading buf before it is reused
    buf ^= 1;
  }

  // C layout: VGPR r -> rows m0+r (lanes 0-15) and m0+r+8 (lanes 16-31)
  for (int r = 0; r < 8; ++r) {
    int row = m0 + r + 8 * half;
    H[(size_t)row * DZ + n0 + l] = sp[row] * acc[r];
  }
}

// -------- cvec[j] = sigma2[j] * sum_i H[i][j]*rinvx[i]  (= P H^T R^-1 x) ---
__global__ void pf_hv_gemv(const float* __restrict__ H,
                           const float* __restrict__ rinvx,
                           const float* __restrict__ sigma2,
                           float* __restrict__ cvec) {
  int j = threadIdx.x;
  float s = 0.f;
  for (int i = 0; i < DX; ++i) s = fmaf(H[(size_t)i * DZ + j], rinvx[i], s);
  cvec[j] = sigma2[j] * s;
}

// -------- WMMA GEMM 2: G = H^T R^-1 H  (128x128, K=784) --------------------
__global__ void pf_gram(const float* __restrict__ H,
                        const float* __restrict__ rinv,
                        float* __restrict__ G) {
  int wid = blockIdx.x * (blockDim.x >> 5) + (threadIdx.x >> 5);
  int lane = threadIdx.x & 31;
  int mt = wid >> 3, nt = wid & 7;           // 8 x 8 tiles
  int m0 = mt * 16, n0 = nt * 16;
  int half = lane >> 4, l = lane & 15;
  v8f acc = {};
#pragma unroll 4
  for (int k = 0; k < DX; k += 4) {
    int ka = k + 2 * half;
    v2f a, b;
    a.x = H[(size_t)ka * DZ + m0 + l] * rinv[ka];       // A[m][ka]=H[ka][m]/r
    a.y = H[(size_t)(ka + 1) * DZ + m0 + l] * rinv[ka + 1];
    b.x = H[(size_t)ka * DZ + n0 + l];                  // B[ka][n]=H[ka][n]
    b.y = H[(size_t)(ka + 1) * DZ + n0 + l];
    acc = wmma_f32_k4(a, b, acc);
  }
  for (int r = 0; r < 8; ++r) {
    int row = m0 + r + 8 * half;
    G[(size_t)row * DZ + n0 + l] = acc[r];
  }
}

// -------- M = G + diag(1/(lam*sigma2)) -------------------------------------
__global__ void pf_form_M(const float* __restrict__ G,
                          const float* __restrict__ sigma2, float lam,
                          float* __restrict__ M) {
  int idx = blockIdx.x * blockDim.x + threadIdx.x;
  if (idx >= DZ * DZ) return;
  int a = idx / DZ, b = idx - a * DZ;
  float v = G[idx];
  if (a == b) v += 1.f / (lam * sigma2[a]);
  M[idx] = v;
}

// -------- in-place Cholesky of 128x128 SPD M (1 block, 128 threads) --------
__global__ void pf_cholesky(float* __restrict__ M) {
  int tid = threadIdx.x;
  for (int k = 0; k < DZ; ++k) {
    __syncthreads();
    if (tid == k) M[k * DZ + k] = sqrtf(M[k * DZ + k]);
    __syncthreads();
    if (tid > k) M[tid * DZ + k] /= M[k * DZ + k];
    __syncthreads();
    if (tid > k) {
      float lik = M[tid * DZ + k];
      for (int j = k + 1; j <= tid; ++j)
        M[tid * DZ + j] -= lik * M[j * DZ + k];
    }
  }
}

// -------- solve (L L^T) X = G, 128 RHS columns (thread per column) ---------
__global__ void pf_trsolve(const float* __restrict__ L,
                           const float* __restrict__ G,
                           float* __restrict__ X) {
  int c = threadIdx.x;
  float y[DZ];
  for (int i = 0; i < DZ; ++i) {
    float s = G[i * DZ + c];
    for (int k = 0; k < i; ++k) s = fmaf(-L[i * DZ + k], y[k], s);
    y[i] = s / L[i * DZ + i];
  }
  for (int i = DZ - 1; i >= 0; --i) {
    float s = y[i];
    for (int k = i + 1; k < DZ; ++k) s = fmaf(-L[k * DZ + i], X[k * DZ + c], s);
    X[i * DZ + c] = s / L[i * DZ + i];
  }
}

// -------- WMMA GEMM 3: Y = G @ X (G symmetric -> coalesced A loads) --------
__global__ void pf_gemm_sym(const float* __restrict__ G,
                            const float* __restrict__ X,
                            float* __restrict__ Y) {
  int wid = blockIdx.x * (blockDim.x >> 5) + (threadIdx.x >> 5);
  int lane = threadIdx.x & 31;
  int mt = wid >> 3, nt = wid & 7;
  int m0 = mt * 16, n0 = nt * 16;
  int half = lane >> 4, l = lane & 15;
  v8f acc = {};
#pragma unroll 4
  for (int k = 0; k < DZ; k += 4) {
    int ka = k + 2 * half;
    v2f a, b;
    a.x = G[(size_t)ka * DZ + m0 + l];        // = G[m][ka] by symmetry
    a.y = G[(size_t)(ka + 1) * DZ + m0 + l];
    b.x = X[(size_t)ka * DZ + n0 + l];
    b.y = X[(size_t)(ka + 1) * DZ + n0 + l];
    acc = wmma_f32_k4(a, b, acc);
  }
  for (int r = 0; r < 8; ++r) {
    int row = m0 + r + 8 * half;
    Y[(size_t)row * DZ + n0 + l] = acc[r];
  }
}

// -------- A = -0.5 * diag(sigma2) * (G - Y)  (Woodbury result) -------------
__global__ void pf_form_A(const float* __restrict__ G,
                          const float* __restrict__ Y,
                          const float* __restrict__ sigma2,
                          float* __restrict__ A) {
  int idx = blockIdx.x * blockDim.x + threadIdx.x;
  if (idx >= DZ * DZ) return;
  int a = idx / DZ;
  A[idx] = -0.5f * sigma2[a] * (G[idx] - Y[idx]);
}

// -------- per-step vector updates (1 block, 128 threads) -------------------
__global__ void pf_step_update(const float* __restrict__ A,
                               const float* __restrict__ cvec,
                               const float* __restrict__ mu,
                               float* __restrict__ bar, float* __restrict__ eta1,
                               float lam, float dl) {
  __shared__ float sc[DZ], smu[DZ], sb[DZ], se[DZ], st[DZ];
  int j = threadIdx.x;
  sc[j] = cvec[j]; smu[j] = mu[j]; sb[j] = bar[j]; se[j] = eta1[j];
  __syncthreads();
  float ac = 0.f, am = 0.f;
  for (int k = 0; k < DZ; ++k) {
    float a = A[j * DZ + k];
    ac = fmaf(a, sc[k], ac);
    am = fmaf(a, smu[k], am);
  }
  float temp = sc[j] + lam * ac + am;        // (I + lam A) c + A mu
  st[j] = temp;
  __syncthreads();
  float at = 0.f, ab = 0.f, ae = 0.f;
  for (int k = 0; k < DZ; ++k) {
    float a = A[j * DZ + k];
    at = fmaf(a, st[k], at);
    ab = fmaf(a, sb[k], ab);
    ae = fmaf(a, se[k], ae);
  }
  float b = temp + 2.f * lam * at;           // (I + 2 lam A) temp
  bar[j] = sb[j] + dl * (ab + b);
  eta1[j] = se[j] + dl * (ae + b);
}

// -------- log|det(I + dl*A)| via Gaussian elimination (1 block) ------------
__global__ void pf_logdet(const float* __restrict__ A, float dl,
                          float* __restrict__ N, float* __restrict__ log_alpha) {
  int tid = threadIdx.x;
  for (int j = 0; j < DZ; ++j)
    N[tid * DZ + j] = dl * A[tid * DZ + j] + (tid == j ? 1.f : 0.f);
  for (int k = 0; k < DZ; ++k) {
    __syncthreads();
    if (tid > k) {
      float f = N[tid * DZ + k] / N[k * DZ + k];
      for (int j = k; j < DZ; ++j) N[tid * DZ + j] -= f * N[k * DZ + j];
    }
  }
  __syncthreads();
  if (tid == 0) {
    float s = 0.f;
    for (int k = 0; k < DZ; ++k) s += logf(fabsf(N[k * DZ + k]));
    *log_alpha += s;
  }
}

// -------- final reduction + output assembly --------------------------------
__global__ void pf_finalize(const float* __restrict__ x,
                            const float* __restrict__ xrec,
                            const float* __restrict__ eta1,
                            const float* __restrict__ z,
                            const float* __restrict__ mu,
                            const float* __restrict__ sigma,
                            const float* __restrict__ log_alpha,
                            float* __restrict__ out) {
  __shared__ float red[256];
  int tid = threadIdx.x;
  float acc = 0.f;
  for (int i = tid; i < DX; i += 256) {
    float xr = xrec[i];
    acc += x[i] * logf(xr) + (1.f - x[i]) * logf(1.f - xr);   // log p(x|z)
    out[i] = xr;
  }
  if (tid < DZ) {
    float e = eta1[tid];
    acc += -0.5f * e * e;                                     // + log_z
    float sg = sigma[tid];
    float dv = z[tid] - mu[tid];
    acc += logf(sg) + 0.5f * dv * dv / (sg * sg);             // - log q(z|x)
    out[DX + tid] = e;
  }
  red[tid] = acc;
  __syncthreads();
  for (int s = 128; s > 0; s >>= 1) {
    if (tid < s) red[tid] += red[tid + s];
    __syncthreads();
  }
  if (tid == 0) {
    float g = *log_alpha;
    out[DX + DZ] = g;              // gamma
    out[DX + DZ + 1] = red[0] + g; // elbo
  }
}

extern "C" void kernel_launch(void* const* d_in, const int* in_sizes, int n_in,
                              void* d_out, int out_size, void* d_ws, size_t ws_size,
                              hipStream_t stream) {
  const float* x    = (const float*)d_in[0];
  const float* eps  = (const float*)d_in[1];
  const float* We1  = (const float*)d_in[2];
  const float* be1  = (const float*)d_in[3];
  const float* Wmu  = (const float*)d_in[4];
  const float* bmu  = (const float*)d_in[5];
  const float* Wsig = (const float*)d_in[6];
  const float* bsig = (const float*)d_in[7];
  const float* Wd1  = (const float*)d_in[8];
  const float* bd1  = (const float*)d_in[9];
  const float* Wd2  = (const float*)d_in[10];
  const float* bd2  = (const float*)d_in[11];
  float* out = (float*)d_out;

  float* ws     = (float*)d_ws;
  float* h      = ws;                               // 8192
  float* tbuf   = h + DH;                           // 8192
  float* dgate  = tbuf + DH;                        // 8192
  float* Wd1T   = dgate + DH;                       // 8192*128
  float* Hm     = Wd1T + (size_t)DH * DZ;           // 784*128
  float* G      = Hm + (size_t)DX * DZ;             // 128*128
  float* Mb     = G + DZ * DZ;
  float* Xb     = Mb + DZ * DZ;
  float* Yb     = Xb + DZ * DZ;
  float* Ab     = Yb + DZ * DZ;
  float* Nb     = Ab + DZ * DZ;
  float* mu     = Nb + DZ * DZ;                     // 128-vectors
  float* sigma  = mu + DZ;
  float* sigma2 = sigma + DZ;
  float* zv     = sigma2 + DZ;
  float* bar    = zv + DZ;
  float* eta1   = bar + DZ;
  float* cvec   = eta1 + DZ;
  float* sp     = cvec + DZ;                        // 784-vectors
  float* rinv   = sp + DX;
  float* rinvx  = rinv + DX;
  float* xrec   = rinvx + DX;
  float* la     = xrec + DX;                        // scalar log_alpha

  // ---- setup ----
  pf_transpose<<<(DH * DZ + 255) / 256, 256, 0, stream>>>(Wd1, Wd1T);
  pf_encode_h<<<DH / 256, 256, 0, stream>>>(x, We1, be1, h);
  pf_encode_muz<<<1, 128, 0, stream>>>(h, Wmu, bmu, Wsig, bsig, eps,
                                       mu, sigma, sigma2, zv, bar, eta1, la);
  pf_dec_hidden<<<DH / 256, 256, 0, stream>>>(zv, Wd1, bd1, tbuf, dgate);
  pf_dec_out<<<(DX + 255) / 256, 256, 0, stream>>>(tbuf, Wd2, bd2, 0, x,
                                                   sp, rinv, rinvx, xrec);

  // ---- 29 particle-flow steps; lambda schedule is compile-time ----
  const double Q = 1.2;
  const int NL = 29;
  const double d1 = (Q - 1.0) / (pow(Q, (double)NL) - 1.0);
  double lam = 0.0;
  for (int s = 0; s < NL; ++s) {
    double dl = d1 * pow(Q, (double)s);
    lam += dl;
    float lamf = (float)lam, dlf = (float)dl;

    pf_dec_hidden<<<DH / 256, 256, 0, stream>>>(bar, Wd1, bd1, tbuf, dgate);
    pf_dec_out<<<(DX + 255) / 256, 256, 0, stream>>>(tbuf, Wd2, bd2, 1, x,
                                                     sp, rinv, rinvx, xrec);
    pf_jacobian<<<49, 256, 0, stream>>>(Wd2, Wd1T, dgate, sp, Hm);
    pf_hv_gemv<<<1, 128, 0, stream>>>(Hm, rinvx, sigma2, cvec);
    pf_gram<<<8, 256, 0, stream>>>(Hm, rinv, G);
    pf_form_M<<<DZ * DZ / 256, 256, 0, stream>>>(G, sigma2, lamf, Mb);
    pf_cholesky<<<1, 128, 0, stream>>>(Mb);
    pf_trsolve<<<1, 128, 0, stream>>>(Mb, G, Xb);
    pf_gemm_sym<<<8, 256, 0, stream>>>(G, Xb, Yb);
    pf_form_A<<<DZ * DZ / 256, 256, 0, stream>>>(G, Yb, sigma2, Ab);
    pf_step_update<<<1, 128, 0, stream>>>(Ab, cvec, mu, bar, eta1, lamf, dlf);
    pf_logdet<<<1, 128, 0, stream>>>(Ab, dlf, Nb, la);
  }

  // ---- final reconstruction + ELBO ----
  pf_dec_hidden<<<DH / 256, 256, 0, stream>>>(eta1, Wd1, bd1, tbuf, dgate);
  pf_dec_out<<<(DX + 255) / 256, 256, 0, stream>>>(tbuf, Wd2, bd2, 2, x,
                                                   sp, rinv, rinvx, xrec);
  pf_finalize<<<1, 256, 0, stream>>>(x, xrec, eta1, zv, mu, sigma, la, out);
}